// NodeTypeConcatSheafLearner_31842887533254
// MI455X (gfx1250) — compile-verified
//
#include <hip/hip_runtime.h>
#include <hip/hip_bf16.h>

typedef __attribute__((ext_vector_type(2))) float v2f;
typedef __attribute__((ext_vector_type(8))) float v8f;

#define NUM_TYPES 4
#define D_IN      32
#define F_TOT     72   // 2*32 + 2*4 = 18 K-chunks of 4
#define OUT_F     9    // 3*3
#define TILES_PER_WAVE 4

// Branch-free tanh from hardware transcendentals:
//   tanh(x) = sign(x) * (1 - 2/(exp(2|x|)+1))
// v_exp_f32 + v_rcp_f32, no EXEC-mask branching (unlike libm tanhf).
__device__ __forceinline__ float fast_tanh(float x) {
    float ax = __builtin_fabsf(x);
    float e  = __expf(2.0f * ax);                              // -> v_exp_f32
    float r  = 1.0f - 2.0f * __builtin_amdgcn_rcpf(e + 1.0f);  // -> v_rcp_f32
    return __builtin_copysignf(r, x);
}

// One wave (32 lanes) handles 16-edge tiles (grid-stride over tiles).
// WMMA mapping:  D[16x16] = A[16x4] x B[4x16] (+0), chained over 18 K-chunks:
//   M (rows of A/D)   = output channel o (0..8, rows 9..15 zero-padded)
//   N (cols of B/D)   = edge within tile (0..15)
//   K chunks 0..7     = x[src] features  0..31
//   K chunks 8..15    = x[dst] features 32..63
//   K chunk  16       = one_hot(type(src))  (B built in registers, no loads)
//   K chunk  17       = one_hot(type(dst))  (B built in registers, no loads)
__global__ __launch_bounds__(256) void sheaf_edge_wmma_kernel(
    const float* __restrict__ x,          // [N, 32]
    const int*   __restrict__ edge_index, // [2, E]
    const int*   __restrict__ node_types, // [N]
    const float* __restrict__ W,          // [9, 72] row-major
    float*       __restrict__ out,        // [E, 9]
    int E)
{
    const int lane = threadIdx.x & 31;
    const int wave = threadIdx.x >> 5;
    const int numTiles   = (E + 15) >> 4;
    const int waveGlobal = blockIdx.x * 8 + wave;
    const int numWaves   = gridDim.x * 8;

    const int col = lane & 15;      // A: row m / B,D: column n (edge)
    const int hi  = lane >> 4;      // lane half selects K sub-pair / D row group
    const int kb  = hi * 2;         // K offset within a chunk of 4

    // ---- A fragments (loop-invariant, hoisted): W[m][4*kc+kb .. +1], zero for m>=9
    v2f afrag[18];
#pragma unroll
    for (int kc = 0; kc < 18; ++kc) {
        v2f a = {0.0f, 0.0f};
        if (col < OUT_F) {
            a = *(const v2f*)(W + col * F_TOT + 4 * kc + kb);  // 8B-aligned
        }
        afrag[kc] = a;
    }

    // ---- wave-uniform grid-stride loop over 16-edge tiles (EXEC all-ones at WMMAs)
    for (int tile = waveGlobal; tile < numTiles; tile += numWaves) {
        // per-lane edge data (edge n = col); clamp tail, mask stores later
        const int  e     = tile * 16 + col;
        const bool valid = (e < E);
        const int  ec    = valid ? e : (E - 1);
        const int  s     = __builtin_nontemporal_load(edge_index + ec);      // streamed once
        const int  d     = __builtin_nontemporal_load(edge_index + E + ec);  // streamed once
        const int  ts    = node_types[s];
        const int  td    = node_types[d];

        // One-hot B fragments for the two type chunks: feature 64+j = (ts==j), 68+j = (td==j)
        v2f bts = { (ts == kb) ? 1.0f : 0.0f, (ts == kb + 1) ? 1.0f : 0.0f };
        v2f btd = { (td == kb) ? 1.0f : 0.0f, (td == kb + 1) ? 1.0f : 0.0f };

        v8f acc = {0.0f, 0.0f, 0.0f, 0.0f, 0.0f, 0.0f, 0.0f, 0.0f};

        // 2 register-only WMMAs for the node-type one-hots
        acc = __builtin_amdgcn_wmma_f32_16x16x4_f32(
            false, afrag[16], false, bts, (short)0, acc, false, false);
        acc = __builtin_amdgcn_wmma_f32_16x16x4_f32(
            false, afrag[17], false, btd, (short)0, acc, false, false);

        // 16 chained f32 WMMAs over K = 64 (src features then dst features)
        const v2f* xs = (const v2f*)(x + (size_t)s * D_IN);
        const v2f* xd = (const v2f*)(x + (size_t)d * D_IN);
#pragma unroll
        for (int kc = 0; kc < 8; ++kc) {
            v2f b = xs[(4 * kc + kb) >> 1];
            acc = __builtin_amdgcn_wmma_f32_16x16x4_f32(
                false, afrag[kc], false, b, (short)0, acc, false, false);
        }
#pragma unroll
        for (int kc = 0; kc < 8; ++kc) {
            v2f b = xd[(4 * kc + kb) >> 1];
            acc = __builtin_amdgcn_wmma_f32_16x16x4_f32(
                false, afrag[8 + kc], false, b, (short)0, acc, false, false);
        }

        // tanh + store (write-only stream -> non-temporal).
        // D layout: lane l, vgpr r holds y[o = r + hi*8][edge col]
        if (valid) {
            float* op = out + (size_t)ec * OUT_F;
            if (hi == 0) {
#pragma unroll
                for (int r = 0; r < 8; ++r)
                    __builtin_nontemporal_store(fast_tanh(acc[r]), op + r);  // o = 0..7
            } else {
                __builtin_nontemporal_store(fast_tanh(acc[0]), op + 8);      // o = 8
            }
        }
    }
}

extern "C" void kernel_launch(void* const* d_in, const int* in_sizes, int n_in,
                              void* d_out, int out_size, void* d_ws, size_t ws_size,
                              hipStream_t stream) {
    const float* x          = (const float*)d_in[0];
    const int*   edge_index = (const int*)d_in[1];
    const int*   node_types = (const int*)d_in[2];
    const float* W          = (const float*)d_in[3];
    float*       out        = (float*)d_out;

    const int E = in_sizes[1] / 2;                      // edge_index is [2, E]
    const int numTiles = (E + 15) / 16;                 // 16-edge tiles, up to TILES_PER_WAVE (4) per wave
    const int wavesNeeded = (numTiles + TILES_PER_WAVE - 1) / TILES_PER_WAVE;
    int blocks = (wavesNeeded + 7) / 8;                 // 8 waves (256 threads) / block
    if (blocks < 1) blocks = 1;

    sheaf_edge_wmma_kernel<<<blocks, 256, 0, stream>>>(x, edge_index, node_types, W, out, E);
}